// TarMACProtocol_18700287607223
// MI455X (gfx1250) — compile-verified
//
#include <hip/hip_runtime.h>

#define NN   8192
#define HID  256
#define MSGD 64
#define KEYD 16
#define LNEPS 1e-5f

typedef __attribute__((ext_vector_type(2))) float v2f;
typedef __attribute__((ext_vector_type(8))) float v8f;

// CDNA5 async global->LDS path (ASYNCcnt-tracked, no VGPR round trip)
#if defined(__gfx1250__) && __has_builtin(__builtin_amdgcn_global_load_async_to_lds_b128)
#define HAS_ASYNC_LDS 1
typedef int v4i __attribute__((vector_size(16)));
typedef __attribute__((address_space(1))) v4i GV4;  // global int4*
typedef __attribute__((address_space(3))) v4i LV4;  // LDS int4*
#else
#define HAS_ASYNC_LDS 0
#endif

__device__ __forceinline__ void wait_asynccnt0() {
#if __has_builtin(__builtin_amdgcn_s_wait_asynccnt)
  __builtin_amdgcn_s_wait_asynccnt(0);
#else
  asm volatile("s_wait_asynccnt 0x0" ::: "memory");
#endif
}

// ---------------- WMMA f32 16x16x4 helpers ----------------
__device__ __forceinline__ v8f wmma4(v2f a, v2f b, v8f c) {
  // D = A(16x4) * B(4x16) + C(16x16), fp32 throughout
  return __builtin_amdgcn_wmma_f32_16x16x4_f32(false, a, false, b, (short)0, c,
                                               false, false);
}

// A-matrix 16x4 f32 layout: lanes 0-15 hold M=0..15; v0 = K=(2*half), v1 = K=(2*half+1)
__device__ __forceinline__ v2f load_a_frag(const float* __restrict__ A, int ld,
                                           int m0, int k0, int lane) {
  int m  = m0 + (lane & 15);
  int kk = k0 + ((lane >> 4) << 1);
  const float* p = A + (size_t)m * ld + kk;
  v2f r; r.x = p[0]; r.y = p[1];
  return r;
}

// B-matrix 4x16 f32 (row-major B[k][n]): row striped across lanes; halves hold K pairs
__device__ __forceinline__ v2f load_b_frag(const float* __restrict__ B, int ld,
                                           int k0, int n0, int lane) {
  int n  = n0 + (lane & 15);
  int kk = k0 + ((lane >> 4) << 1);
  v2f r; r.x = B[(size_t)kk * ld + n]; r.y = B[(size_t)(kk + 1) * ld + n];
  return r;
}

// B[k][n] = W[n][k] for weights stored (out,in) row-major (h @ W^T)
__device__ __forceinline__ v2f load_bT_frag(const float* __restrict__ W, int ld,
                                            int k0, int n0, int lane) {
  int n  = n0 + (lane & 15);
  int kk = k0 + ((lane >> 4) << 1);
  const float* p = W + (size_t)n * ld + kk;
  v2f r; r.x = p[0]; r.y = p[1];
  return r;
}

// C/D 16x16 f32: VGPR v: lanes0-15 -> M=v, lanes16-31 -> M=v+8; col = lane&15
__device__ __forceinline__ void store_d(float* __restrict__ D, int ld, int m0,
                                        int n0, int lane, v8f d) {
  int col = n0 + (lane & 15);
  int rb  = m0 + ((lane >> 4) << 3);
#pragma unroll
  for (int v = 0; v < 8; ++v) D[(size_t)(rb + v) * ld + col] = d[v];
}

// ---------------- K1: msgs/keys/qrys projections ----------------
__global__ void __launch_bounds__(128)
proj_kernel(const float* __restrict__ h,
            const float* __restrict__ Wm, const float* __restrict__ bm,
            const float* __restrict__ Wk, const float* __restrict__ bk,
            const float* __restrict__ Wq, const float* __restrict__ bq,
            float* __restrict__ msgs, float* __restrict__ keys,
            float* __restrict__ qrys) {
  int lane = threadIdx.x & 31;
  int wid  = blockIdx.x * 4 + (threadIdx.x >> 5);
  int rt   = wid / 6;   // row tile
  int t    = wid % 6;   // 0..3 msgs tiles, 4 keys, 5 qrys  (wave-uniform branch)
  int m0   = rt * 16;
  const float *W, *b; float* out; int n0, ldo;
  if (t < 4)       { W = Wm; b = bm; out = msgs; n0 = t * 16; ldo = MSGD; }
  else if (t == 4) { W = Wk; b = bk; out = keys; n0 = 0;      ldo = KEYD; }
  else             { W = Wq; b = bq; out = qrys; n0 = 0;      ldo = KEYD; }

  v8f acc = {};
  for (int k0 = 0; k0 < HID; k0 += 4) {
    v2f a  = load_a_frag(h, HID, m0, k0, lane);
    v2f bb = load_bT_frag(W, HID, k0, n0, lane);
    acc = wmma4(a, bb, acc);
  }
  int col   = n0 + (lane & 15);
  int rb    = m0 + ((lane >> 4) << 3);
  float bia = b[col];
#pragma unroll
  for (int v = 0; v < 8; ++v) out[(size_t)(rb + v) * ldo + col] = acc[v] + bia;
}

// ---------------- K2: raw scores S = (q @ k^T) / sqrt(KEY) ----------------
__global__ void __launch_bounds__(256)
score_kernel(const float* __restrict__ qrys, const float* __restrict__ keys,
             float* __restrict__ S) {
  int lane = threadIdx.x & 31;
  int wid  = blockIdx.x * 8 + (threadIdx.x >> 5);
  int it   = wid >> 9;        // NN/16 = 512 tiles per row
  int jt   = wid & 511;
  int i0 = it * 16, j0 = jt * 16;
  v8f acc = {};
#pragma unroll
  for (int k0 = 0; k0 < KEYD; k0 += 4) {
    v2f a = load_a_frag(qrys, KEYD, i0, k0, lane);
    v2f b = load_bT_frag(keys, KEYD, k0, j0, lane);
    acc = wmma4(a, b, acc);
  }
  const float scale = 0.25f;  // 1/sqrt(16)
#pragma unroll
  for (int v = 0; v < 8; ++v) acc[v] *= scale;
  store_d(S, NN, i0, j0, lane, acc);
}

// ---------------- K3: row softmax in place ----------------
__global__ void __launch_bounds__(256)
softmax_kernel(float* __restrict__ S) {
  __shared__ float redm[8];
  __shared__ float reds[8];
  int tid  = threadIdx.x;
  float* p = S + (size_t)blockIdx.x * NN;

  float v[32];
  float m = -3.402823466e38f;
#pragma unroll
  for (int i = 0; i < 32; ++i) { v[i] = p[tid + 256 * i]; m = fmaxf(m, v[i]); }
#pragma unroll
  for (int off = 16; off; off >>= 1) m = fmaxf(m, __shfl_xor(m, off, 32));
  if ((tid & 31) == 0) redm[tid >> 5] = m;
  __syncthreads();
  float bm = redm[0];
#pragma unroll
  for (int i = 1; i < 8; ++i) bm = fmaxf(bm, redm[i]);

  float s = 0.f;
#pragma unroll
  for (int i = 0; i < 32; ++i) { v[i] = __expf(v[i] - bm); s += v[i]; }
#pragma unroll
  for (int off = 16; off; off >>= 1) s += __shfl_xor(s, off, 32);
  if ((tid & 31) == 0) reds[tid >> 5] = s;
  __syncthreads();
  float bs = 0.f;
#pragma unroll
  for (int i = 0; i < 8; ++i) bs += reds[i];
  float inv = 1.f / bs;
#pragma unroll
  for (int i = 0; i < 32; ++i) p[tid + 256 * i] = v[i] * inv;
}

// ---------------- K4: agg = attn @ msgs (K = 8192, async-LDS staged A) ----------------
#define AGG_KC 128
__global__ void __launch_bounds__(128)
agg_kernel(const float* __restrict__ attn, const float* __restrict__ msgs,
           float* __restrict__ agg) {
  __shared__ float lds[64 * AGG_KC];  // 32 KB: 64 rows x 128 k-cols of attn
  int tid  = threadIdx.x;
  int lane = tid & 31;
  int wave = tid >> 5;
  int r0   = blockIdx.x * 64;  // block owns 64 rows, 16 per wave
  int m0   = wave * 16;

  v8f acc0 = {}, acc1 = {}, acc2 = {}, acc3 = {};
  for (int kc = 0; kc < NN; kc += AGG_KC) {
    __syncthreads();
#pragma unroll
    for (int idx = tid; idx < 64 * (AGG_KC / 4); idx += 128) {
      int rr = idx >> 5;            // AGG_KC/4 == 32 float4 per row
      int cc = (idx & 31) << 2;
      const float* src = &attn[(size_t)(r0 + rr) * NN + kc + cc];
      float* dst = &lds[rr * AGG_KC + cc];
#if HAS_ASYNC_LDS
      // global -> LDS DMA, tracked by ASYNCcnt (no VGPR round trip)
      __builtin_amdgcn_global_load_async_to_lds_b128((GV4*)src, (LV4*)dst, 0, 0);
#else
      *(float4*)dst = *(const float4*)src;
#endif
      if (kc + AGG_KC < NN) __builtin_prefetch(src + AGG_KC, 0, 0);
    }
#if HAS_ASYNC_LDS
    wait_asynccnt0();
#endif
    __syncthreads();
#pragma unroll 4
    for (int k0 = 0; k0 < AGG_KC; k0 += 4) {
      v2f a = load_a_frag(lds, AGG_KC, m0, k0, lane);
      acc0 = wmma4(a, load_b_frag(msgs, MSGD, kc + k0, 0,  lane), acc0);
      acc1 = wmma4(a, load_b_frag(msgs, MSGD, kc + k0, 16, lane), acc1);
      acc2 = wmma4(a, load_b_frag(msgs, MSGD, kc + k0, 32, lane), acc2);
      acc3 = wmma4(a, load_b_frag(msgs, MSGD, kc + k0, 48, lane), acc3);
    }
  }
  store_d(agg, MSGD, r0 + m0, 0,  lane, acc0);
  store_d(agg, MSGD, r0 + m0, 16, lane, acc1);
  store_d(agg, MSGD, r0 + m0, 32, lane, acc2);
  store_d(agg, MSGD, r0 + m0, 48, lane, acc3);
}

// ---------------- K5: z = [h|agg] @ W_agg^T + b ; LayerNorm ; ReLU ----------------
__global__ void __launch_bounds__(512)
fuse_kernel(const float* __restrict__ h, const float* __restrict__ agg,
            const float* __restrict__ Wa, const float* __restrict__ ba,
            const float* __restrict__ lng, const float* __restrict__ lnb,
            float* __restrict__ hout) {
  __shared__ float hbuf[16 * HID];   // 16 KB
  __shared__ float abuf[16 * MSGD];  //  4 KB
  __shared__ float zbuf[16 * HID];   // 16 KB
  int tid = threadIdx.x, lane = tid & 31, wave = tid >> 5;
  int r0 = blockIdx.x * 16;

  for (int idx = tid; idx < 16 * (HID / 4); idx += 512) {
    int rr = idx / (HID / 4), cc = (idx % (HID / 4)) * 4;
    *(float4*)&hbuf[rr * HID + cc] =
        *(const float4*)&h[(size_t)(r0 + rr) * HID + cc];
  }
  for (int idx = tid; idx < 16 * (MSGD / 4); idx += 512) {
    int rr = idx / (MSGD / 4), cc = (idx % (MSGD / 4)) * 4;
    *(float4*)&abuf[rr * MSGD + cc] =
        *(const float4*)&agg[(size_t)(r0 + rr) * MSGD + cc];
  }
  __syncthreads();

  int n0 = wave * 16;                // 16 waves cover 256 output cols
  v8f acc = {};
  for (int k0 = 0; k0 < HID; k0 += 4) {
    v2f a = load_a_frag(hbuf, HID, 0, k0, lane);
    v2f b = load_bT_frag(Wa, HID + MSGD, k0, n0, lane);
    acc = wmma4(a, b, acc);
  }
#pragma unroll
  for (int k0 = 0; k0 < MSGD; k0 += 4) {
    v2f a = load_a_frag(abuf, MSGD, 0, k0, lane);
    v2f b = load_bT_frag(Wa + HID, HID + MSGD, k0, n0, lane);  // W[:, 256+k]
    acc = wmma4(a, b, acc);
  }
  {
    int col = n0 + (lane & 15);
    int rb  = (lane >> 4) << 3;
    float bia = ba[col];
#pragma unroll
    for (int v = 0; v < 8; ++v) zbuf[(rb + v) * HID + col] = acc[v] + bia;
  }
  __syncthreads();

  // LayerNorm + ReLU: wave w normalizes row w (256 values, 8 per lane)
  const float* zr = &zbuf[wave * HID];
  float x[8];
  float s = 0.f;
#pragma unroll
  for (int i = 0; i < 8; ++i) { x[i] = zr[lane + 32 * i]; s += x[i]; }
#pragma unroll
  for (int off = 16; off; off >>= 1) s += __shfl_xor(s, off, 32);
  float mu = s * (1.0f / HID);
  float vv = 0.f;
#pragma unroll
  for (int i = 0; i < 8; ++i) { float d = x[i] - mu; vv += d * d; }
#pragma unroll
  for (int off = 16; off; off >>= 1) vv += __shfl_xor(vv, off, 32);
  float rinv = rsqrtf(vv * (1.0f / HID) + LNEPS);
  float* orow = hout + (size_t)(r0 + wave) * HID;
#pragma unroll
  for (int i = 0; i < 8; ++i) {
    int c = lane + 32 * i;
    float y = lng[c] * (x[i] - mu) * rinv + lnb[c];
    orow[c] = fmaxf(y, 0.f);
  }
}

// ---------------- host launcher ----------------
extern "C" void kernel_launch(void* const* d_in, const int* in_sizes, int n_in,
                              void* d_out, int out_size, void* d_ws,
                              size_t ws_size, hipStream_t stream) {
  const float* hs = (const float*)d_in[0];
  const float* Wm = (const float*)d_in[1];
  const float* bm = (const float*)d_in[2];
  const float* Wk = (const float*)d_in[3];
  const float* bk = (const float*)d_in[4];
  const float* Wq = (const float*)d_in[5];
  const float* bq = (const float*)d_in[6];
  const float* Wa = (const float*)d_in[7];
  const float* ba = (const float*)d_in[8];
  const float* lg = (const float*)d_in[9];
  const float* lb = (const float*)d_in[10];

  float* out_h = (float*)d_out;                 // (N, HID)
  float* S     = out_h + (size_t)NN * HID;      // (N, N) attn region (scratch+final)

  char* w = (char*)d_ws;
  float* h1   = (float*)w; w += sizeof(float) * NN * HID;
  float* msgs = (float*)w; w += sizeof(float) * NN * MSGD;
  float* keys = (float*)w; w += sizeof(float) * NN * KEYD;
  float* qrys = (float*)w; w += sizeof(float) * NN * KEYD;
  float* agg  = (float*)w; w += sizeof(float) * NN * MSGD;

  const float* hin = hs;
  for (int r = 0; r < 2; ++r) {
    float* hout = (r == 0) ? h1 : out_h;
    proj_kernel<<<(NN / 16) * 6 / 4, 128, 0, stream>>>(hin, Wm, bm, Wk, bk, Wq,
                                                       bq, msgs, keys, qrys);
    score_kernel<<<(NN / 16) * (NN / 16) / 8, 256, 0, stream>>>(qrys, keys, S);
    softmax_kernel<<<NN, 256, 0, stream>>>(S);
    agg_kernel<<<NN / 64, 128, 0, stream>>>(S, msgs, agg);
    fuse_kernel<<<NN / 16, 512, 0, stream>>>(hin, agg, Wa, ba, lg, lb, hout);
    hin = h1;
  }
}